// IntraVoxelConvTorch_8375186227641
// MI455X (gfx1250) — compile-verified
//
#include <hip/hip_runtime.h>
#include <hip/hip_bf16.h>
#include <math.h>

typedef __attribute__((ext_vector_type(2))) float v2f;
typedef __attribute__((ext_vector_type(8))) float v8f;

#define C_IN   16
#define V_DIM  45
#define DIN    48
#define DOUT   46
#define O_DIM  32
#define KTOT   432        // 27 * 16, k = kidx*16 + c
#define KSTEPS 108        // 432 / 4
#define WTILE  24         // output w' per workgroup (3 columns per wave)
#define WIN    26         // staged input w per workgroup (WTILE + 2)
#define WINP   27         // LDS stride (odd -> conflict-free b64 B loads)
#define NWCH   2          // chunks: [0,24), [24,46) (22 cols, tail masked)

// ---------------------------------------------------------------------------
// Kernel 1: fused logmap + 3x3x3 conv as f32 WMMA GEMM (M=32, K=432, N tiles
// of 16 v-components per output voxel). Writes pre-expmap y into d_out.
// ---------------------------------------------------------------------------
__global__ __launch_bounds__(256)
void conv_wmma_kernel(const float* __restrict__ x,
                      const float* __restrict__ wt,
                      float* __restrict__ y) {
  // B slab: v2f index ((dh*8 + c/2)*WINP + w)*16 + n ; .x/.y = c even/odd.
  __shared__ __align__(16) float sB[9 * 8 * WINP * 16 * 2];   // 248832 B
  // A fragments in exact 16x16x4 lane layout: [kk][o-tile][lane] = float2.
  __shared__ __align__(16) v2f sA[KSTEPS * 2 * 32];           //  55296 B

  const int tid   = threadIdx.x;
  const int vt    = blockIdx.x % 3;   // v tile (0,16,32)
  const int wch   = blockIdx.x / 3;   // w' chunk
  const int hp    = blockIdx.y;       // h'
  const int dp    = blockIdx.z;       // d'
  const int v0    = vt * 16;
  const int wbase = wch * WTILE;

  // ---- stage A: swizzle weights into WMMA A-fragment layout ----
  for (int u = tid; u < KSTEPS * 2 * 32; u += 256) {
    int kk   = u >> 6;
    int rem  = u & 63;
    int lane = rem & 31;
    int t    = rem >> 5;
    int m    = t * 16 + (lane & 15);
    int half = lane >> 4;
    int k0   = kk * 4 + 2 * half;
    v2f a;
    a.x = wt[m * KTOT + k0];
    a.y = wt[m * KTOT + k0 + 1];
    sA[u] = a;
  }

  // ---- stage B: logmap(x) slab; one acos/sqrt per 4-element unit ----
  const int npts = 9 * C_IN * WIN;    // (dh, c, w) points
  for (int u = tid; u < npts * 4; u += 256) {
    int vq   = u & 3;
    int p    = u >> 2;
    int wl   = p % WIN;
    int rest = p / WIN;
    int c    = rest & 15;
    int dh   = rest >> 4;              // 0..8 = kd*3 + kh
    int d  = dp + dh / 3;
    int h  = hp + (dh - (dh / 3) * 3);
    int wg = wbase + wl;
    bool wvalid = (wg < DIN);          // last chunk overhangs; pad zeros
    int gbase = (((c * DIN + d) * DIN + h) * DIN + wg) * V_DIM;
    float x0 = wvalid ? x[gbase] : 1.0f;
    float c0 = fminf(fmaxf(x0, -1.0f + 1e-6f), 1.0f - 1e-6f);
    float theta = acosf(c0);
    float sn = sqrtf(fmaxf(1.0f - c0 * c0, 1e-30f));  // sin(acos(c0))
    float f = theta / sn;
    int cp = c >> 1, par = c & 1;
    int fb = (((dh * 8 + cp) * WINP + wl) * 16) * 2 + par;
#pragma unroll
    for (int i = 0; i < 4; ++i) {
      int nn = vq * 4 + i;
      int v  = v0 + nn;
      float val = 0.0f;
      if (wvalid && v < V_DIM) {
        float xv = x[gbase + v];
        val = (v == 0) ? f * (xv - c0) : f * xv;
      }
      sB[fb + nn * 2] = val;
    }
  }
  __syncthreads();

  // ---- main WMMA loop: 3 columns/wave, shared A frags, additive addressing
  const int lane = tid & 31;
  const int wave = tid >> 5;
  const int half = lane >> 4;
  const int nn   = lane & 15;
  const int wl0  = wave;
  const int wl1  = wave + 8;
  const int wl2  = wave + 16;
  const int wp0  = wbase + wl0;
  const int wp1  = wbase + wl1;
  const int wp2  = wbase + wl2;

  v8f acc00 = {0.f,0.f,0.f,0.f,0.f,0.f,0.f,0.f};
  v8f acc01 = {0.f,0.f,0.f,0.f,0.f,0.f,0.f,0.f};
  v8f acc10 = {0.f,0.f,0.f,0.f,0.f,0.f,0.f,0.f};
  v8f acc11 = {0.f,0.f,0.f,0.f,0.f,0.f,0.f,0.f};
  v8f acc20 = {0.f,0.f,0.f,0.f,0.f,0.f,0.f,0.f};
  v8f acc21 = {0.f,0.f,0.f,0.f,0.f,0.f,0.f,0.f};

  const int lanebase = half * (WINP * 16) + nn;   // c-parity half selects K row pair
  int aoff = lane;                                 // v2f index into sA, +64 per step

#pragma unroll 1
  for (int kd = 0; kd < 3; ++kd) {
#pragma unroll 1
    for (int kh = 0; kh < 3; ++kh) {
      const int dhterm = (kd * 3 + kh) * (8 * WINP * 16);
#pragma unroll 1
      for (int kw = 0; kw < 3; ++kw) {
        int b0 = dhterm + (wl0 + kw) * 16 + lanebase;
        int b1 = dhterm + (wl1 + kw) * 16 + lanebase;
        int b2 = dhterm + (wl2 + kw) * 16 + lanebase;
#pragma unroll
        for (int c4 = 0; c4 < 4; ++c4) {
          v2f bb0 = *(const v2f*)&sB[(b0 + c4 * (2 * WINP * 16)) * 2];
          v2f bb1 = *(const v2f*)&sB[(b1 + c4 * (2 * WINP * 16)) * 2];
          v2f bb2 = *(const v2f*)&sB[(b2 + c4 * (2 * WINP * 16)) * 2];
          v2f a0  = sA[aoff];
          v2f a1  = sA[aoff + 32];
          acc00 = __builtin_amdgcn_wmma_f32_16x16x4_f32(false, a0, false, bb0,
                                                        (short)0, acc00, false, false);
          acc01 = __builtin_amdgcn_wmma_f32_16x16x4_f32(false, a1, false, bb0,
                                                        (short)0, acc01, false, false);
          acc10 = __builtin_amdgcn_wmma_f32_16x16x4_f32(false, a0, false, bb1,
                                                        (short)0, acc10, false, false);
          acc11 = __builtin_amdgcn_wmma_f32_16x16x4_f32(false, a1, false, bb1,
                                                        (short)0, acc11, false, false);
          acc20 = __builtin_amdgcn_wmma_f32_16x16x4_f32(false, a0, false, bb2,
                                                        (short)0, acc20, false, false);
          acc21 = __builtin_amdgcn_wmma_f32_16x16x4_f32(false, a1, false, bb2,
                                                        (short)0, acc21, false, false);
          aoff += 64;
        }
      }
    }
  }

  // ---- epilogue: D element (M = q + 8*half, N = nn) in VGPR q ----
  const int v = v0 + nn;
  const int ostride = DOUT * DOUT * DOUT * V_DIM;  // 4,380,120
  if (v < V_DIM) {
    if (wp0 < DOUT) {
      int base = ((dp * DOUT + hp) * DOUT + wp0) * V_DIM + v;
#pragma unroll
      for (int q = 0; q < 8; ++q) {
        int o = q + 8 * half;
        y[base + o * ostride]        = acc00[q];
        y[base + (o + 16) * ostride] = acc01[q];
      }
    }
    if (wp1 < DOUT) {
      int base = ((dp * DOUT + hp) * DOUT + wp1) * V_DIM + v;
#pragma unroll
      for (int q = 0; q < 8; ++q) {
        int o = q + 8 * half;
        y[base + o * ostride]        = acc10[q];
        y[base + (o + 16) * ostride] = acc11[q];
      }
    }
    if (wp2 < DOUT) {
      int base = ((dp * DOUT + hp) * DOUT + wp2) * V_DIM + v;
#pragma unroll
      for (int q = 0; q < 8; ++q) {
        int o = q + 8 * half;
        y[base + o * ostride]        = acc20[q];
        y[base + (o + 16) * ostride] = acc21[q];
      }
    }
  }
}

// ---------------------------------------------------------------------------
// Kernel 2: in-place expmap over the 45-vector of each (o, d', h', w') point.
// LDS-staged for coalesced global access; 45-stride is odd -> conflict-free.
// ---------------------------------------------------------------------------
__global__ __launch_bounds__(256)
void expmap_kernel(float* __restrict__ y) {
  __shared__ float s[256 * V_DIM];
  const int tid = threadIdx.x;
  const int base = blockIdx.x * (256 * V_DIM);
  for (int i = tid; i < 256 * V_DIM; i += 256) s[i] = y[base + i];
  __syncthreads();
  float* v = &s[tid * V_DIM];
  float ss = 0.0f;
#pragma unroll
  for (int i = 0; i < V_DIM; ++i) ss += v[i] * v[i];
  float nrm  = sqrtf(ss);
  float safe = fmaxf(nrm, 1e-12f);
  float f    = sinf(safe) / safe;
  float cn   = cosf(nrm);
#pragma unroll
  for (int i = 0; i < V_DIM; ++i) v[i] *= f;
  v[0] += cn;
  __syncthreads();
  for (int i = tid; i < 256 * V_DIM; i += 256) y[base + i] = s[i];
}

extern "C" void kernel_launch(void* const* d_in, const int* in_sizes, int n_in,
                              void* d_out, int out_size, void* d_ws, size_t ws_size,
                              hipStream_t stream) {
  const float* x  = (const float*)d_in[0];   // (1,16,48,48,48,45) f32
  const float* wt = (const float*)d_in[1];   // (32, 432) f32
  float* out = (float*)d_out;                // (1,32,46,46,46,45) f32

  dim3 g1(NWCH * 3, DOUT, DOUT);             // (wchunk*vtile, h', d')
  conv_wmma_kernel<<<g1, 256, 0, stream>>>(x, wt, out);

  int npoints = O_DIM * DOUT * DOUT * DOUT;  // 3,114,752 = 12167 * 256 exactly
  expmap_kernel<<<npoints / 256, 256, 0, stream>>>(out);
}